// SparseMax_8091718386028
// MI455X (gfx1250) — compile-verified
//
#include <hip/hip_runtime.h>
#include <stdint.h>

// ---------------------------------------------------------------------------
// Sparsemax over last axis (D=512) via Michelot's exact simplex projection.
// One wave32 per row; row lives in 16 VGPRs/lane; rows stream through LDS
// using CDNA5 async global->LDS copies (ASYNCcnt) with double buffering.
// ---------------------------------------------------------------------------

#ifndef __has_builtin
#define __has_builtin(x) 0
#endif

#define AS1 __attribute__((address_space(1)))
#define AS3 __attribute__((address_space(3)))

typedef int v4i __attribute__((vector_size(16)));

constexpr int kD    = 512;            // sparsemax axis length
constexpr int kWave = 32;             // CDNA5 is wave32-only
constexpr int kFPL  = kD / kWave;     // 16 floats held per lane
constexpr int kWPB  = 8;              // waves per block (256 threads)

// ---- CDNA5 async global->LDS 16B copy (tracked by ASYNCcnt) ----
__device__ __forceinline__ void async_copy_b128(const float* gsrc, float* lds_dst) {
#if __has_builtin(__builtin_amdgcn_global_load_async_to_lds_b128)
  __builtin_amdgcn_global_load_async_to_lds_b128(
      (AS1 v4i*)(uintptr_t)gsrc,                       // global src (as1)
      (AS3 v4i*)(uint32_t)(uintptr_t)lds_dst,          // LDS dst: low 32 bits of
      0, 0);                                           // generic ptr = LDS addr
#else
  uint32_t lds_off = (uint32_t)(uintptr_t)lds_dst;
  uint64_t gaddr   = (uint64_t)(uintptr_t)gsrc;
  asm volatile("global_load_async_to_lds_b128 %0, %1, off"
               :: "v"(lds_off), "v"(gaddr) : "memory");
#endif
}

#if __has_builtin(__builtin_amdgcn_s_wait_asynccnt)
#define WAIT_ASYNC(n) do { __builtin_amdgcn_s_wait_asynccnt(n); \
                           asm volatile("" ::: "memory"); } while (0)
#else
#define WAIT_ASYNC(n) asm volatile("s_wait_asynccnt %0" :: "n"(n) : "memory")
#endif

// ---- wave32 butterfly reductions (every lane gets the total) ----
__device__ __forceinline__ float wave_sum_f32(float v) {
#pragma unroll
  for (int m = 16; m >= 1; m >>= 1) v += __shfl_xor(v, m, kWave);
  return v;
}
__device__ __forceinline__ int wave_sum_i32(int v) {
#pragma unroll
  for (int m = 16; m >= 1; m >>= 1) v += __shfl_xor(v, m, kWave);
  return v;
}

__global__ __launch_bounds__(kWPB * kWave)
void sparsemax_michelot_kernel(const float* __restrict__ in,
                               float* __restrict__ out,
                               int rows, int total_waves) {
  // 8 waves * 2 row-buffers * 2KB = 32KB static LDS per block
  __shared__ __align__(16) float smem[kWPB * 2 * kD];

  const int lane = threadIdx.x & (kWave - 1);
  const int wib  = threadIdx.x / kWave;              // wave index in block
  const int w    = blockIdx.x * kWPB + wib;          // global wave id

  float* const buf0 = &smem[(wib * 2 + 0) * kD];
  float* const buf1 = &smem[(wib * 2 + 1) * kD];

  int row = w;
  if (row < rows) {
    const float* src = in + (size_t)row * kD;
#pragma unroll
    for (int j = 0; j < 4; ++j)
      async_copy_b128(src + j * 128 + lane * 4, buf0 + j * 128 + lane * 4);
  }

  int cur = 0;
  for (; row < rows; row += total_waves) {
    // Kick off next row's copy before touching the current one.
    const int nxt = row + total_waves;
    if (nxt < rows) {
      float* nb = (cur == 0) ? buf1 : buf0;
      const float* src = in + (size_t)nxt * kD;
#pragma unroll
      for (int j = 0; j < 4; ++j)
        async_copy_b128(src + j * 128 + lane * 4, nb + j * 128 + lane * 4);
      WAIT_ASYNC(4);   // async loads retire in order: current row's 4 are done
    } else {
      WAIT_ASYNC(0);
    }

    // Pull this wave's row from LDS into registers (16 floats/lane).
    const float* b = (cur == 0) ? buf0 : buf1;
    float f[kFPL];
#pragma unroll
    for (int j = 0; j < 4; ++j) {
      float4 v = *(const float4*)(b + j * 128 + lane * 4);
      f[4 * j + 0] = v.x; f[4 * j + 1] = v.y;
      f[4 * j + 2] = v.z; f[4 * j + 3] = v.w;
    }

    // ---- Michelot iteration: exact sparsemax threshold, no sort ----
    float ps = 0.f;
#pragma unroll
    for (int k = 0; k < kFPL; ++k) ps += f[k];
    float tau   = (wave_sum_f32(ps) - 1.0f) * (1.0f / (float)kD);
    int   cprev = kD;

    for (int it = 0; it < 32; ++it) {
      float s = 0.f; int c = 0;
#pragma unroll
      for (int k = 0; k < kFPL; ++k) {
        bool a = f[k] > tau;
        s += a ? f[k] : 0.f;
        c += a ? 1 : 0;
      }
      float S = wave_sum_f32(s);
      int   C = wave_sum_i32(c);     // C >= 1 always (max(z) > tau invariant)
      tau = (S - 1.0f) / (float)C;
      if (C == cprev) break;         // support stabilized -> tau is exact
      cprev = C;
    }

    // ---- weights = relu(z - tau), streamed out as b128 ----
    float* dst = out + (size_t)row * kD;
#pragma unroll
    for (int j = 0; j < 4; ++j) {
      float4 o;
      o.x = fmaxf(f[4 * j + 0] - tau, 0.f);
      o.y = fmaxf(f[4 * j + 1] - tau, 0.f);
      o.z = fmaxf(f[4 * j + 2] - tau, 0.f);
      o.w = fmaxf(f[4 * j + 3] - tau, 0.f);
      *(float4*)(dst + j * 128 + lane * 4) = o;
    }
    cur ^= 1;
  }
}

extern "C" void kernel_launch(void* const* d_in, const int* in_sizes, int n_in,
                              void* d_out, int out_size, void* d_ws, size_t ws_size,
                              hipStream_t stream) {
  const float* in  = (const float*)d_in[0];   // inputs [B,N,D] fp32 (mask unused)
  float*       out = (float*)d_out;

  const int rows = in_sizes[0] / kD;          // B*N rows of length D
  int blocks = (rows + kWPB - 1) / kWPB;
  if (blocks > 1024) blocks = 1024;           // 8192 waves -> 4 rows/wave pipeline
  const int total_waves = blocks * kWPB;

  sparsemax_michelot_kernel<<<dim3(blocks), dim3(kWPB * kWave), 0, stream>>>(
      in, out, rows, total_waves);
}